// DilateLoss_34634616275209
// MI455X (gfx1250) — compile-verified
//
#include <hip/hip_runtime.h>

// DILATE loss (soft-DTW value + path gradient) for B=128, N=128, F=8, fp32.
// One workgroup (128 threads = 4 wave32) per (b,f) problem; R and E tables in
// LDS (~130 KB/WG -- needs gfx1250's 320KB-per-WGP LDS). Anti-diagonal
// wavefront DP forward + backward with register-carried same-lane parents,
// then a branchless WMMA f32 16x16x4 contraction E * [1, k, k^2] for the
// Omega-weighted temporal term.

#define GAMMA_F  0.01f
#define BIG_F    1e8f
#define NN       128
#define NFEAT    8
#define NPROB    1024   // B*F

typedef __attribute__((ext_vector_type(2))) float v2f;
typedef __attribute__((ext_vector_type(8))) float v8f;

#define LDS_FLOATS (2*NN*NN + NN + NN + 128)

extern "C" __global__ void __launch_bounds__(128, 1)
sdtw_dp_kernel(const float* __restrict__ outputs,
               const float* __restrict__ targets,
               float* __restrict__ partials)
{
  extern __shared__ float smem[];
  float* Rm  = smem;              // R[1..N][1..N], 0-based storage, 128x128
  float* Em  = smem + NN*NN;      // E (dSDTW/dD), 128x128
  float* tv  = smem + 2*NN*NN;    // targets column (128)
  float* ov  = tv + NN;           // outputs column (128)
  float* red = ov + NN;           // block-reduction scratch (128)

  const int p   = blockIdx.x;     // problem id = b*8 + f
  const int b   = p >> 3;
  const int f   = p & 7;
  const int tid = threadIdx.x;    // 0..127, lane owns column j = tid
  const int j   = tid;

  tv[tid] = targets[(b*NN + tid)*NFEAT + f];
  ov[tid] = outputs[(b*NN + tid)*NFEAT + f];
  __syncthreads();

  const float g  = GAMMA_F;
  const float ig = 1.0f / GAMMA_F;
  const float ov_j = ov[j];

  // ---------------- forward DP: R[i][j] = D + softmin_g(diag, up, left) ----
  // "up" parent R[i-1][j] is this lane's previous result -> carried register.
  float u_reg = BIG_F;
  for (int d = 0; d < 2*NN - 1; ++d) {
    const int i = d - j;
    if (0 <= i && i < NN) {
      const float dt  = tv[i] - ov_j;
      const float Dij = dt * dt;
      const float a = (i > 0 && j > 0) ? Rm[(i-1)*NN + (j-1)]
                      : ((i == 0 && j == 0) ? 0.0f : BIG_F);
      const float u = u_reg;
      const float l = (j > 0) ? Rm[i*NN + (j-1)] : BIG_F;
      const float m = fminf(a, fminf(u, l));
      // stable softmin: exponents <= 0; exp((m-BIG)/g) underflows to 0 cleanly
      const float s = __expf((m - a)*ig) + __expf((m - u)*ig) + __expf((m - l)*ig);
      const float r = Dij + m - g*__logf(s);
      Rm[i*NN + j] = r;
      u_reg = r;
    }
    __syncthreads();
  }

  // ---------------- backward DP: E[i][j] (path / gradient) -----------------
  // E[i][j] = sum over children c of E[c] * exp((R[c] - D[c] - R[i][j])/g)
  // Down child (i+1,j) is this lane's previous backward cell -> carry E, R, t.
  const float ov_j1 = (j < NN-1) ? ov[j+1] : 0.0f;
  float e_down = 0.0f, r_down = 0.0f, t_ip1 = 0.0f;
  for (int d = 2*NN - 2; d >= 0; --d) {
    const int i = d - j;
    if (0 <= i && i < NN) {
      const float rij = Rm[i*NN + j];
      const float t_i = tv[i];
      float e;
      if (i == NN-1 && j == NN-1) {
        e = 1.0f;
      } else {
        e = 0.0f;
        if (i < NN-1) {                       // child (i+1, j): same lane
          const float dt = t_ip1 - ov_j;
          e += __expf((r_down - dt*dt - rij)*ig) * e_down;
        }
        if (j < NN-1) {                       // child (i, j+1): lane j+1
          const float dt = t_i - ov_j1;
          e += __expf((Rm[i*NN + (j+1)] - dt*dt - rij)*ig) * Em[i*NN + (j+1)];
          if (i < NN-1) {                     // child (i+1, j+1): lane j+1
            const float dt2 = t_ip1 - ov_j1;
            e += __expf((Rm[(i+1)*NN + (j+1)] - dt2*dt2 - rij)*ig)
                 * Em[(i+1)*NN + (j+1)];
          }
        }
      }
      Em[i*NN + j] = e;
      e_down = e;
      r_down = rij;
      t_ip1  = t_i;
    }
    __syncthreads();
  }

  // -------- WMMA f32 16x16x4: moments (r_i, m_i, q_i) = E * [1, k, k^2] ----
  // sum_ij E[i][j]*(i-j)^2 = sum_i ( i^2*r_i - 2*i*m_i + q_i )
  const int wave = tid >> 5;
  const int lane = tid & 31;
  const int h    = lane >> 4;     // half-wave select (K sub-block, ISA layout)
  const int n    = lane & 15;     // tile column (and A-row index)
  // branchless per-lane selection weights: column n of B holds k^n for n<3
  const float sel0 = (n == 0) ? 1.0f : 0.0f;
  const float sel1 = (n == 1) ? 1.0f : 0.0f;
  const float sel2 = (n == 2) ? 1.0f : 0.0f;
  const float m2s1 = -2.0f * sel1;
  float partial = 0.0f;

  for (int mb = wave; mb < 8; mb += 4) {      // each wave: 2 blocks of 16 rows
    v8f c = {};
    const float* Arow = Em + (16*mb + n)*NN;  // A: M = lane % 16
    for (int kc = 0; kc < 32; ++kc) {         // K in chunks of 4
      const int klo = 4*kc + 2*h;             // A: lanes 0-15 K=0,1; 16-31 K=2,3
      v2f av, bv;
      av.x = Arow[klo];
      av.y = Arow[klo + 1];
      const float k0 = (float)klo, k1 = k0 + 1.0f;
      bv.x = fmaf(k0, fmaf(k0, sel2, sel1), sel0);   // sel0 + sel1*k + sel2*k^2
      bv.y = fmaf(k1, fmaf(k1, sel2, sel1), sel0);
      c = __builtin_amdgcn_wmma_f32_16x16x4_f32(false, av, false, bv,
                                                (short)0, c, false, false);
    }
    // C/D layout: VGPR r of lane -> row M = r + 8*h (in tile), col N = lane%16
#pragma unroll
    for (int r = 0; r < 8; ++r) {
      const float fi = (float)(16*mb + 8*h + r);       // global row index i
      const float coef = fmaf(fi, fmaf(fi, sel0, m2s1), sel2);
      partial = fmaf(coef, c[r], partial);
    }
  }

  red[tid] = partial;
  __syncthreads();
  for (int off = 64; off > 0; off >>= 1) {
    if (tid < off) red[tid] += red[tid + off];
    __syncthreads();
  }
  if (tid == 0) {
    partials[2*p]     = Rm[(NN-1)*NN + (NN-1)];   // SDTW value R[N,N]
    partials[2*p + 1] = red[0];                   // sum E*Omega
  }
}

extern "C" __global__ void __launch_bounds__(256)
sdtw_reduce_kernel(const float* __restrict__ partials, float* __restrict__ out)
{
  __shared__ float sv[256], ss[256];
  const int t = threadIdx.x;
  float v = 0.0f, s = 0.0f;
  for (int k = t; k < NPROB; k += 256) { v += partials[2*k]; s += partials[2*k + 1]; }
  sv[t] = v; ss[t] = s;
  __syncthreads();
  for (int off = 128; off > 0; off >>= 1) {
    if (t < off) { sv[t] += sv[t + off]; ss[t] += ss[t + off]; }
    __syncthreads();
  }
  if (t == 0) {
    const float loss_shape    = sv[0] * (1.0f / (float)NPROB);
    const float loss_temporal = ss[0] * (1.0f / ((float)NPROB * (float)NN * (float)NN));
    out[0] = 0.5f * loss_shape + 0.5f * loss_temporal;  // ALPHA = 0.5
    out[1] = loss_shape;
    out[2] = loss_temporal;
  }
}

extern "C" void kernel_launch(void* const* d_in, const int* in_sizes, int n_in,
                              void* d_out, int out_size, void* d_ws, size_t ws_size,
                              hipStream_t stream) {
  (void)in_sizes; (void)n_in; (void)out_size; (void)ws_size;
  const float* outputs = (const float*)d_in[0];
  const float* targets = (const float*)d_in[1];
  float* partials = (float*)d_ws;          // 1024 x 2 floats
  float* out      = (float*)d_out;         // [loss, loss_shape, loss_temporal]

  const size_t lds_bytes = LDS_FLOATS * sizeof(float);  // ~130 KB (gfx1250: 320KB/WGP)
  (void)hipFuncSetAttribute((const void*)sdtw_dp_kernel,
                            hipFuncAttributeMaxDynamicSharedMemorySize,
                            (int)lds_bytes);

  sdtw_dp_kernel<<<NPROB, 128, lds_bytes, stream>>>(outputs, targets, partials);
  sdtw_reduce_kernel<<<1, 256, 0, stream>>>(partials, out);
}